// MultiHeadAttention_12283606467769
// MI455X (gfx1250) — compile-verified
//
#include <hip/hip_runtime.h>
#include <hip/hip_bf16.h>

typedef __attribute__((ext_vector_type(2))) float v2f;
typedef __attribute__((ext_vector_type(8))) float v8f;

#define NB     8
#define SEQ    1024
#define EMBED  1024
#define NHEADS 16
#define DH     64
#define SROW   1028   // padded LDS row stride (floats) for the score strip

__device__ __forceinline__ v8f wmma4(v2f a, v2f b, v8f c) {
    // V_WMMA_F32_16X16X4_F32 : D = A(16x4) * B(4x16) + C(16x16), full fp32
    return __builtin_amdgcn_wmma_f32_16x16x4_f32(
        /*neg_a=*/false, a, /*neg_b=*/false, b,
        /*c_mod=*/(short)0, c, /*reuse_a=*/false, /*reuse_b=*/false);
}

// ---------------------------------------------------------------------------
// Kernel 1: per-head projections  Xp = X_head @ W^T   (W is (out,in) = (64,64))
//   grid.x = NB*NHEADS*(SEQ/16) / 4 tiles, 4 waves/block, one 16-row tile/wave
//   grid.y = 0:V 1:K 2:Q
//   output layout: [n][h][l][64]
// ---------------------------------------------------------------------------
__global__ __launch_bounds__(128) void mha_proj_kernel(
    const float* __restrict__ V, const float* __restrict__ K,
    const float* __restrict__ Q,
    const float* __restrict__ Wv, const float* __restrict__ Wk,
    const float* __restrict__ Wq,
    float* __restrict__ Vp, float* __restrict__ Kp, float* __restrict__ Qp)
{
    const int which = blockIdx.y;
    const float* __restrict__ x = (which == 0) ? V : (which == 1) ? K : Q;
    const float* __restrict__ W = (which == 0) ? Wv : (which == 1) ? Wk : Wq;
    float* __restrict__ out     = (which == 0) ? Vp : (which == 1) ? Kp : Qp;

    const int wave = threadIdx.x >> 5;
    const int lane = threadIdx.x & 31;
    const int mrow = lane & 15;          // M row within tile
    const int kk   = (lane >> 4) << 1;   // K sub-offset: 0 (lanes 0-15) or 2

    const int tile = blockIdx.x * 4 + wave;         // 0 .. 8191
    const int lt   = tile & 63;                     // 16-token tile within seq
    const int nh   = tile >> 6;                     // n*NHEADS + h
    const int h    = nh & (NHEADS - 1);
    const int n    = nh >> 4;
    const int l0   = lt * 16;

    // A fragments: rows of the input head slice, 16 K-steps of 4
    const float* xrow = x + ((size_t)(n * SEQ + l0 + mrow)) * EMBED + h * DH;
    v2f a[16];
#pragma unroll
    for (int ks = 0; ks < 16; ++ks)
        a[ks] = *(const v2f*)(xrow + 4 * ks + kk);

    float* obase = out + ((size_t)nh * SEQ + l0) * DH;
#pragma unroll
    for (int nt = 0; nt < 4; ++nt) {
        const int ncol = nt * 16 + mrow;           // output feature (row of W)
        const float* wrow = W + ncol * DH;
        v8f c = {0.f,0.f,0.f,0.f,0.f,0.f,0.f,0.f};
#pragma unroll
        for (int ks = 0; ks < 16; ++ks) {
            v2f b = *(const v2f*)(wrow + 4 * ks + kk);   // B[k][n] = W[n][k]
            c = wmma4(a[ks], b, c);
        }
#pragma unroll
        for (int v = 0; v < 8; ++v) {
            const int m = v + 8 * (lane >> 4);
            obase[(size_t)m * DH + ncol] = c[v];
        }
    }
}

// ---------------------------------------------------------------------------
// Kernel 2: fused attention per (n, h, 16-row q-tile)
//   8 waves / block.  Score strip 16x1024 lives in LDS (padded stride 1028).
//   attention tensor is NT-stored (537 MB >> 192 MB L2).
// ---------------------------------------------------------------------------
__global__ __launch_bounds__(256) void mha_attn_kernel(
    const float* __restrict__ Qp, const float* __restrict__ Kp,
    const float* __restrict__ Vp, const int* __restrict__ mask,
    float* __restrict__ attn, float* __restrict__ Oc)
{
    extern __shared__ float lds[];
    float* S      = lds;                 // 16*SROW  score/prob strip
    float* red    = S + 16 * SROW;       // 256      partial max/sum
    float* rowmax = red + 256;           // 16
    float* rowsum = rowmax + 16;         // 16
    float* Cpart  = rowsum + 16;         // 4*256    O partial-tile reduce

    const int nh = blockIdx.x;                       // n*NHEADS + h
    const int qt = blockIdx.y;                       // q tile 0..63
    const int h  = nh & (NHEADS - 1);
    const int n  = nh >> 4;

    const int tid  = threadIdx.x;
    const int w    = tid >> 5;
    const int lane = tid & 31;
    const int mrow = lane & 15;
    const int kk   = (lane >> 4) << 1;
    const int hi   = lane >> 4;

    const float* Qbase = Qp + ((size_t)nh * SEQ + qt * 16) * DH;
    const float* Kbase = Kp + (size_t)nh * SEQ * DH;
    const float* Vbase = Vp + (size_t)nh * SEQ * DH;

    // Q A-fragments (16 rows x 64), identical in all waves (small, L2-cached)
    v2f qa[16];
#pragma unroll
    for (int ks = 0; ks < 16; ++ks)
        qa[ks] = *(const v2f*)(Qbase + (size_t)mrow * DH + 4 * ks + kk);

    const float scale = 0.03125f;        // 1/sqrt(EMBED) = 1/32

    // ---- S = Q K^T * scale, masked; wave w covers k-tiles w, w+8, ...
    for (int i = 0; i < 8; ++i) {
        const int kt = w + 8 * i;
        const float* krow = Kbase + (size_t)(kt * 16 + mrow) * DH;
        v8f c = {0.f,0.f,0.f,0.f,0.f,0.f,0.f,0.f};
#pragma unroll
        for (int ks = 0; ks < 16; ++ks) {
            v2f b = *(const v2f*)(krow + 4 * ks + kk);   // B[k][n] = Kp[col][k]
            c = wmma4(qa[ks], b, c);
        }
        const int col  = kt * 16 + mrow;
        const int mval = mask[n * SEQ + col];
#pragma unroll
        for (int v = 0; v < 8; ++v) {
            const int m = v + 8 * hi;
            float s = c[v] * scale;
            if (mval == 0) s = -1.0e20f;
            S[m * SROW + col] = s;
        }
    }
    __syncthreads();

    // ---- softmax over each of the 16 rows (16 threads per row)
    const int r = tid >> 4;
    const int g = tid & 15;
    float mx = -3.0e38f;
    for (int i = g; i < SEQ; i += 16) mx = fmaxf(mx, S[r * SROW + i]);
    red[r * 16 + g] = mx;
    __syncthreads();
    if (tid < 16) {
        float m2 = red[tid * 16];
#pragma unroll
        for (int i = 1; i < 16; ++i) m2 = fmaxf(m2, red[tid * 16 + i]);
        rowmax[tid] = m2;
    }
    __syncthreads();
    const float rm = rowmax[r];
    float sum = 0.f;
    for (int i = g; i < SEQ; i += 16) {
        float e = __expf(S[r * SROW + i] - rm);
        S[r * SROW + i] = e;
        sum += e;
    }
    red[r * 16 + g] = sum;
    __syncthreads();
    if (tid < 16) {
        float s2 = 0.f;
#pragma unroll
        for (int i = 0; i < 16; ++i) s2 += red[tid * 16 + i];
        rowsum[tid] = s2;
    }
    __syncthreads();
    const float inv = 1.0f / rowsum[r];
    float* arow = attn + ((size_t)nh * SEQ + qt * 16 + r) * SEQ;
    for (int i = g; i < SEQ; i += 16) {
        float p = S[r * SROW + i] * inv;
        S[r * SROW + i] = p;
        __builtin_nontemporal_store(p, arow + i);   // huge tensor: bypass cache
    }
    __syncthreads();

    // ---- O = P(16x1024) @ V(1024x64); wave = (ntile 0..3, khalf 0..1)
    const int ntile = w & 3;
    const int khalf = w >> 2;
    const int ncol  = ntile * 16 + mrow;
    v8f oc = {0.f,0.f,0.f,0.f,0.f,0.f,0.f,0.f};
    for (int ks = 0; ks < 128; ++ks) {
        const int kb = khalf * 512 + ks * 4;
        v2f a = *(const v2f*)(&S[mrow * SROW + kb + kk]);       // A from LDS
        const float* vrow = Vbase + (size_t)(kb + kk) * DH + ncol;
        v2f b;
        b.x = vrow[0];
        b.y = vrow[DH];
        oc = wmma4(a, b, oc);
    }
    if (khalf == 1) {
#pragma unroll
        for (int v = 0; v < 8; ++v)
            Cpart[ntile * 256 + (v + 8 * hi) * 16 + mrow] = oc[v];
    }
    __syncthreads();
    if (khalf == 0) {
        float* obase = Oc + ((size_t)(n * SEQ + qt * 16)) * EMBED + h * DH;
#pragma unroll
        for (int v = 0; v < 8; ++v) {
            const int m = v + 8 * hi;
            obase[(size_t)m * EMBED + ncol] =
                oc[v] + Cpart[ntile * 256 + m * 16 + mrow];
        }
    }
}

// ---------------------------------------------------------------------------
// Kernel 3: out = Oc @ W_out^T + b_out     (8192x1024) x (1024x1024)
//   8 waves share a 16-row A tile staged through LDS in 256-col chunks.
// ---------------------------------------------------------------------------
__global__ __launch_bounds__(256) void mha_outproj_kernel(
    const float* __restrict__ Oc, const float* __restrict__ Wout,
    const float* __restrict__ bout, float* __restrict__ out)
{
    __shared__ float Alds[16 * 260];     // padded stride 260

    const int tid  = threadIdx.x;
    const int w    = tid >> 5;
    const int lane = tid & 31;
    const int mrow = lane & 15;
    const int kk   = (lane >> 4) << 1;
    const int hi   = lane >> 4;

    const int mtile = blockIdx.x;                     // 0..511
    const int ncol  = (blockIdx.y * 8 + w) * 16 + mrow;
    const float* brow = Wout + (size_t)ncol * EMBED;

    v8f c = {0.f,0.f,0.f,0.f,0.f,0.f,0.f,0.f};
    for (int cc = 0; cc < 4; ++cc) {
        // cooperative, coalesced stage of A[16][256] chunk
        const float* abase = Oc + ((size_t)mtile * 16) * EMBED + cc * 256;
        for (int idx = tid; idx < 4096; idx += 256)
            Alds[(idx >> 8) * 260 + (idx & 255)] = abase[(size_t)(idx >> 8) * EMBED + (idx & 255)];
        __syncthreads();
#pragma unroll 4
        for (int ks = 0; ks < 64; ++ks) {
            v2f a = *(const v2f*)(&Alds[mrow * 260 + 4 * ks + kk]);
            v2f b = *(const v2f*)(brow + cc * 256 + 4 * ks + kk);
            c = wmma4(a, b, c);
        }
        __syncthreads();
    }
    const float bb = bout[ncol];
    float* obase = out + (size_t)mtile * 16 * EMBED;
#pragma unroll
    for (int v = 0; v < 8; ++v) {
        const int m = v + 8 * hi;
        obase[(size_t)m * EMBED + ncol] = c[v] + bb;
    }
}

// ---------------------------------------------------------------------------
extern "C" void kernel_launch(void* const* d_in, const int* in_sizes, int n_in,
                              void* d_out, int out_size, void* d_ws, size_t ws_size,
                              hipStream_t stream) {
    (void)in_sizes; (void)n_in; (void)out_size; (void)ws_size;
    const float* V    = (const float*)d_in[0];
    const float* K    = (const float*)d_in[1];
    const float* Q    = (const float*)d_in[2];
    const int*   mask = (const int*)d_in[3];
    const float* Wv   = (const float*)d_in[4];
    const float* Wk   = (const float*)d_in[5];
    const float* Wq   = (const float*)d_in[6];
    const float* Wout = (const float*)d_in[7];
    const float* bout = (const float*)d_in[8];

    float* out  = (float*)d_out;                               // (N,L,E)
    float* attn = out + (size_t)NB * SEQ * EMBED;              // (N,H,L,L)

    float* ws = (float*)d_ws;
    const size_t per = (size_t)NB * NHEADS * SEQ * DH;         // 8.39M floats
    float* Vp = ws;
    float* Kp = ws + per;
    float* Qp = ws + 2 * per;
    float* Oc = ws + 3 * per;                                  // (N,L,E) concat heads

    // 1) projections: 8192 tiles x 3 matrices, 4 waves/block
    dim3 g1(NB * NHEADS * (SEQ / 16) / 4, 3);
    mha_proj_kernel<<<g1, 128, 0, stream>>>(V, K, Q, Wv, Wk, Wq, Vp, Kp, Qp);

    // 2) fused attention: (n*h, q-tile) grid; ~71 KB dynamic LDS (<= 320 KB/WGP)
    const size_t smem = (size_t)(16 * SROW + 256 + 32 + 4 * 256) * sizeof(float);
    dim3 g2(NB * NHEADS, SEQ / 16);
    mha_attn_kernel<<<g2, 256, smem, stream>>>(Qp, Kp, Vp, mask, attn, Oc);

    // 3) output projection
    dim3 g3((NB * SEQ) / 16, EMBED / (16 * 8));
    mha_outproj_kernel<<<g3, 256, 0, stream>>>(Oc, Wout, bout, out);
}